// BaseModel_53549652247037
// MI455X (gfx1250) — compile-verified
//
#include <hip/hip_runtime.h>

typedef __attribute__((ext_vector_type(2))) float v2f;
typedef __attribute__((ext_vector_type(8))) float v8f;

#define TBINS 50
#define TBW   0.02f
#define TN    8192
#define TD    32
#define TE    100000
#define TS    256
#define TEPS  1e-6f
#define EVT_BLOCKS 512

__device__ __forceinline__ float waveReduceSum(float x) {
  #pragma unroll
  for (int off = 16; off > 0; off >>= 1) x += __shfl_xor(x, off, 32);
  return x;
}

// ---------------- Event log-intensity term ----------------
// One wave per event, lane = dimension (coalesced 128B per node access).
// Means cancel in all differences -> use raw x0 / v directly.
__global__ __launch_bounds__(256) void event_kernel(
    const float* __restrict__ x0, const float* __restrict__ v,
    const float* __restrict__ beta, const float* __restrict__ et,
    const int* __restrict__ pairs, float* __restrict__ partial) {
  const int lane = threadIdx.x & 31;
  const int wid  = threadIdx.x >> 5;
  const int gwave = blockIdx.x * 8 + wid;
  const int nwaves = gridDim.x * 8;
  float acc = 0.0f;
  for (int e = gwave; e < TE; e += nwaves) {
    const int p0 = pairs[e];
    const int p1 = pairs[TE + e];
    const float t  = et[e];
    const float fb = floorf(t / TBW);          // matches jnp floor(t / BW)
    int b = (int)fb; b = b < 0 ? 0 : (b > TBINS - 1 ? TBINS - 1 : b);
    const float res = t - fb * TBW;            // jnp remainder semantics
    float dxt = x0[p0 * TD + lane] - x0[p1 * TD + lane];
    for (int k = 0; k < b; ++k) {
      const float* vk = v + (size_t)k * TN * TD;
      dxt += TBW * (vk[p0 * TD + lane] - vk[p1 * TD + lane]);
    }
    const float* vb = v + (size_t)b * TN * TD;
    const float dv = vb[p0 * TD + lane] - vb[p1 * TD + lane];
    dxt += res * dv;
    const float s = waveReduceSum(dxt * dxt);
    if (lane == 0) acc += -s + beta[p0] + beta[p1];
  }
  __shared__ float redw[8];
  if (lane == 0) redw[wid] = acc;
  __syncthreads();
  if (threadIdx.x == 0) {
    float s = 0.0f;
    #pragma unroll
    for (int w = 0; w < 8; ++w) s += redw[w];
    partial[blockIdx.x] = s;
  }
}

// ---------------- Pairwise survival-integral term ----------------
// One workgroup per bin. Gram matrices of P_b (positions) and W_b (velocities)
// over the 256 selected nodes via V_WMMA_F32_16X16X4_F32 (exact f32 MACs).
__global__ __launch_bounds__(256) void pair_kernel(
    const float* __restrict__ x0, const float* __restrict__ v,
    const float* __restrict__ beta, const int* __restrict__ nodes,
    float* __restrict__ partial) {
  const int b   = blockIdx.x;        // bin
  const int tid = threadIdx.x;
  const int lane = tid & 31, wid = tid >> 5;
  const int lm = lane & 15, lh = lane >> 4;

  __shared__ float Pl[TS * 33];                 // padded stride 33 (bank spread)
  __shared__ float ppd[TS], vvd[TS], pvd[TS], betaS[TS];
  __shared__ float red[256];

  const float* vb = v + (size_t)b * TN * TD;

  // Phase 1: positions at lower bin bound t=b*BW into LDS.
  for (int idx = tid; idx < TS * TD; idx += 256) {
    const int n = idx >> 5, d = idx & 31;
    const int node = nodes[n];
    float p = x0[node * TD + d];
    for (int k = 0; k < b; ++k)
      p += TBW * v[(size_t)k * TN * TD + node * TD + d];
    Pl[n * 33 + d] = p;
  }
  __syncthreads();
  // Phase 1b: diagonals ||P||^2, ||W||^2, P.W and beta per node.
  if (tid < TS) {
    const int node = nodes[tid];
    float spp = 0.0f, svv = 0.0f, spv = 0.0f;
    for (int d = 0; d < TD; ++d) {
      const float pw = Pl[tid * 33 + d];
      const float w  = vb[node * TD + d];
      spp += pw * pw; svv += w * w; spv += pw * w;
    }
    ppd[tid] = spp; vvd[tid] = svv; pvd[tid] = spv;
    betaS[tid] = beta[node];
  }
  __syncthreads();

  const float tl = (float)b * TBW, tu = (float)(b + 1) * TBW;
  const float half_sqrt_pi = 0.88622692545f;
  float acc = 0.0f;

  // 136 upper-triangular 16x16 tiles; 8 waves x 17 tiles each (uniform EXEC).
  for (int u = wid; u < 136; u += 8) {
    int ti = 0, rem = u;
    while (rem >= 16 - ti) { rem -= 16 - ti; ++ti; }
    const int tj = ti + rem;
    const int rowI = ti * 16 + lm, rowJ = tj * 16 + lm;
    const int nodeI = nodes[rowI], nodeJ = nodes[rowJ];

    v8f dpp = {}, dvv = {}, dpv = {}, dvp = {};
    #pragma unroll
    for (int kk = 0; kk < 8; ++kk) {
      const int col = kk * 4 + lh * 2;          // per-lane K columns
      v2f aP = { Pl[rowI * 33 + col], Pl[rowI * 33 + col + 1] };
      v2f bP = { Pl[rowJ * 33 + col], Pl[rowJ * 33 + col + 1] };
      v2f aW = { vb[nodeI * TD + col], vb[nodeI * TD + col + 1] };
      v2f bW = { vb[nodeJ * TD + col], vb[nodeJ * TD + col + 1] };
      dpp = __builtin_amdgcn_wmma_f32_16x16x4_f32(false, aP, false, bP, (short)0, dpp, false, false);
      dvv = __builtin_amdgcn_wmma_f32_16x16x4_f32(false, aW, false, bW, (short)0, dvv, false, false);
      dpv = __builtin_amdgcn_wmma_f32_16x16x4_f32(false, aP, false, bW, (short)0, dpv, false, false);
      dvp = __builtin_amdgcn_wmma_f32_16x16x4_f32(false, aW, false, bP, (short)0, dvp, false, false);
    }

    const int j = tj * 16 + lm;                 // C/D: N = lane%16
    #pragma unroll
    for (int r = 0; r < 8; ++r) {
      const int i = ti * 16 + r + 8 * lh;       // C/D: M = r + 8*(lane/16)
      if (i < j) {
        const float ndXt2 = fmaxf(ppd[i] + ppd[j] - 2.0f * dpp[r], 0.0f);
        const float ndV2  = fmaxf(vvd[i] + vvd[j] - 2.0f * dvv[r], 0.0f);
        const float dotxv = pvd[i] + pvd[j] - dpv[r] - dvp[r];
        const float ndV = sqrtf(ndV2);
        const float inv = 1.0f / (ndV + TEPS);
        const float rr  = dotxv * inv;
        const float term0 = half_sqrt_pi * inv;
        const float term1 = expf(betaS[i] + betaS[j] + rr * rr - ndXt2);
        const float ediff = erff(tu * ndV + rr) - erff(tl * ndV + rr);
        acc += term0 * term1 * ediff;
      }
    }
  }

  red[tid] = acc;
  __syncthreads();
  #pragma unroll
  for (int s = 128; s > 0; s >>= 1) {
    if (tid < s) red[tid] += red[tid + s];
    __syncthreads();
  }
  if (tid == 0) partial[b] = red[0];
}

// ---------------- Final deterministic combine ----------------
__global__ void final_kernel(const float* __restrict__ wsE,
                             const float* __restrict__ wsP,
                             float* __restrict__ out) {
  if (threadIdx.x == 0 && blockIdx.x == 0) {
    double se = 0.0, sp = 0.0;
    for (int i = 0; i < EVT_BLOCKS; ++i) se += (double)wsE[i];
    for (int i = 0; i < TBINS; ++i)      sp += (double)wsP[i];
    out[0] = (float)(sp - se);
  }
}

extern "C" void kernel_launch(void* const* d_in, const int* in_sizes, int n_in,
                              void* d_out, int out_size, void* d_ws, size_t ws_size,
                              hipStream_t stream) {
  (void)in_sizes; (void)n_in; (void)out_size; (void)ws_size;
  const float* x0    = (const float*)d_in[0];
  const float* v     = (const float*)d_in[1];
  const float* beta  = (const float*)d_in[2];
  const float* et    = (const float*)d_in[3];
  const int*   pairs = (const int*)d_in[4];
  const int*   nodes = (const int*)d_in[5];
  float* wsE = (float*)d_ws;            // EVT_BLOCKS partials
  float* wsP = wsE + EVT_BLOCKS;        // TBINS partials

  event_kernel<<<EVT_BLOCKS, 256, 0, stream>>>(x0, v, beta, et, pairs, wsE);
  pair_kernel<<<TBINS, 256, 0, stream>>>(x0, v, beta, nodes, wsP);
  final_kernel<<<1, 32, 0, stream>>>(wsE, wsP, (float*)d_out);
}